// ModelNew_1580547972719
// MI455X (gfx1250) — compile-verified
//
#include <hip/hip_runtime.h>
#include <hip/hip_bf16.h>

// MI455X (gfx1250) fused QKV projection + causal-ReLU attention.
// All three GEMM stages run on V_WMMA_F32_16X16X4_F32 (wave32 matrix op).
// Global->LDS staging uses the CDNA5 async-to-LDS path (ASYNCcnt) with
// double-buffered tiles in the GEMM so the copy overlaps the matrix pipe.
// GEMM wave tile is 32Mx64N (2x4 WMMA accumulators) to cut LDS fragment
// traffic to 0.75 loads/WMMA (LDS bandwidth is the likely fp32-WMMA limiter).
//
// Shapes: B=2, T=2048, C=2048, H=16, hs=128, 3C=6144, M=B*T=4096.
// d_ws holds proj (M x 3C f32 = ~100.7 MB) -- fits in the 192MB L2.

typedef __attribute__((ext_vector_type(2))) float v2f;
typedef __attribute__((ext_vector_type(8))) float v8f;

#define C_EMBD   2048
#define C3       6144
#define N_HEAD   16
#define HS       128
#define BATCH    2
#define TLEN     2048
#define MROWS    (BATCH * TLEN)   // 4096

// ---- CDNA5 async global->LDS (writes LDS directly, tracked by ASYNCcnt) ----
// Flat/generic pointers to LDS carry the DS offset in their low 32 bits
// (via the LDS aperture), which is exactly what the
// instruction's VDST operand wants.
__device__ __forceinline__ void async_ld_b128(const void* lds_dst, const void* gsrc) {
    uint32_t ldsoff = (uint32_t)(uintptr_t)lds_dst;
    asm volatile("global_load_async_to_lds_b128 %0, %1, off"
                 :: "v"(ldsoff), "v"(gsrc) : "memory");
}
__device__ __forceinline__ void wait_async0() {
    asm volatile("s_wait_asynccnt 0x0" ::: "memory");
}

// ---------------------------------------------------------------------------
// Kernel 1: proj[M][3C] = X[M][C] * W[3C][C]^T + bias
// Block: 128 threads (4 waves). Block tile 128(M) x 64(N). Wave: 32 rows x 64
// cols = 2x4 accumulators of 16x16 (64 acc VGPRs). K staged 16/iter through
// DOUBLE-BUFFERED LDS filled by async-to-LDS; fill of tile i+1 overlaps WMMA
// on tile i. Per k-step: 2 A-frags + 4 B-frags feed 8 WMMAs.
// ---------------------------------------------------------------------------
__global__ __launch_bounds__(128) void qkv_gemm_wmma(
    const float* __restrict__ X, const float* __restrict__ W,
    const float* __restrict__ bias, float* __restrict__ P)
{
    __shared__ float As[2][128][20];  // X tile [m][k], padded rows (80B stride)
    __shared__ float Bs[2][64][20];   // W tile [n][k]

    const int tid  = threadIdx.x;
    const int lane = tid & 31;
    const int wave = tid >> 5;        // 0..3
    const int l16  = lane & 15;
    const int sub  = lane >> 4;       // 0/1 : which half of the wave
    const int m0   = blockIdx.x * 128;
    const int n0   = blockIdx.y * 64;

    // This thread's 16B chunks per tile: A rows r0+{0,32,64,96}, B rows r0+{0,32}
    const int r0 = tid >> 2;
    const int c0 = (tid & 3) * 4;
    const float* gA0 = &X[(size_t)(m0 + r0)      * C_EMBD + c0];
    const float* gA1 = &X[(size_t)(m0 + r0 + 32) * C_EMBD + c0];
    const float* gA2 = &X[(size_t)(m0 + r0 + 64) * C_EMBD + c0];
    const float* gA3 = &X[(size_t)(m0 + r0 + 96) * C_EMBD + c0];
    const float* gB0 = &W[(size_t)(n0 + r0)      * C_EMBD + c0];
    const float* gB1 = &W[(size_t)(n0 + r0 + 32) * C_EMBD + c0];

    v8f acc[2][4] = {};

    // Prologue: async-fill buffer 0 with the k0=0 tile.
    async_ld_b128(&As[0][r0][c0],      gA0);
    async_ld_b128(&As[0][r0 + 32][c0], gA1);
    async_ld_b128(&As[0][r0 + 64][c0], gA2);
    async_ld_b128(&As[0][r0 + 96][c0], gA3);
    async_ld_b128(&Bs[0][r0][c0],      gB0);
    async_ld_b128(&Bs[0][r0 + 32][c0], gB1);
    wait_async0();
    __syncthreads();

    for (int k0 = 0; k0 < C_EMBD; k0 += 16) {
        const int cur = (k0 >> 4) & 1;
        const int nxt = cur ^ 1;

        // Kick off the next tile's async fill; it runs under the WMMA below.
        if (k0 + 16 < C_EMBD) {
            const int kn = k0 + 16;
            async_ld_b128(&As[nxt][r0][c0],      gA0 + kn);
            async_ld_b128(&As[nxt][r0 + 32][c0], gA1 + kn);
            async_ld_b128(&As[nxt][r0 + 64][c0], gA2 + kn);
            async_ld_b128(&As[nxt][r0 + 96][c0], gA3 + kn);
            async_ld_b128(&Bs[nxt][r0][c0],      gB0 + kn);
            async_ld_b128(&Bs[nxt][r0 + 32][c0], gB1 + kn);
        }

        #pragma unroll
        for (int kk = 0; kk < 16; kk += 4) {
            // A frags: lane holds A[M=l16][K = kk + 2*sub + {0,1}]
            v2f a0 = *(const v2f*)&As[cur][wave * 32 + l16][kk + 2 * sub];
            v2f a1 = *(const v2f*)&As[cur][wave * 32 + 16 + l16][kk + 2 * sub];
            #pragma unroll
            for (int nt = 0; nt < 4; ++nt) {
                // B frag: lane holds B[K = kk+2*sub+{0,1}][N=l16] = W[n][k]
                v2f b = *(const v2f*)&Bs[cur][nt * 16 + l16][kk + 2 * sub];
                acc[0][nt] = __builtin_amdgcn_wmma_f32_16x16x4_f32(
                    false, a0, false, b, (short)0, acc[0][nt], false, false);
                acc[1][nt] = __builtin_amdgcn_wmma_f32_16x16x4_f32(
                    false, a1, false, b, (short)0, acc[1][nt], false, false);
            }
        }

        // My next-tile async writes have landed; barrier publishes them to the
        // block and (via its dscnt wait) retires this tile's ds reads before
        // the buffer is overwritten next iteration.
        wait_async0();
        __syncthreads();
    }

    // Epilogue: bias add + store. D layout: M = vr + 8*sub, N = l16.
    #pragma unroll
    for (int nt = 0; nt < 4; ++nt) {
        const int n  = n0 + nt * 16 + l16;
        const float bv = bias[n];
        #pragma unroll
        for (int mt = 0; mt < 2; ++mt) {
            #pragma unroll
            for (int vr = 0; vr < 8; ++vr) {
                const int m = m0 + wave * 32 + mt * 16 + vr + 8 * sub;
                P[(size_t)m * C3 + n] = acc[mt][nt][vr] + bv;
            }
        }
    }
}

// ---------------------------------------------------------------------------
// Kernel 2: causal ReLU attention, streaming (no softmax => no rescaling).
// Grid: (T/64, H, B). Block: 128 threads (4 waves); wave w owns q rows
// [q0+16w, q0+16w+16). Per key tile of 16: S = Q*K^T (32 WMMA), mask, then
// Y += S*V (32 WMMA). Q and K tiles are filled with async-to-LDS; V is
// staged transposed (through VGPRs) so its B-fragments are contiguous
// ds_load_b64; S re-layouts D->A through per-wave LDS scratch.
// ---------------------------------------------------------------------------
__global__ __launch_bounds__(128) void relu_attn_wmma(
    const float* __restrict__ P, float* __restrict__ Y)
{
    __shared__ float Qs[4][16][HS + 4];   // per-wave 16 q-rows x 128
    __shared__ float Ks[16][HS + 4];      // key tile [key][d]
    __shared__ float Vt[HS][18];          // value tile transposed [d][key]
    __shared__ float Ss[4][16][18];       // per-wave masked score tile [m][n]

    const int tid  = threadIdx.x;
    const int lane = tid & 31;
    const int wave = tid >> 5;
    const int l16  = lane & 15;
    const int sub  = lane >> 4;
    const int q0   = blockIdx.x * 64;
    const int h    = blockIdx.y;
    const int b    = blockIdx.z;
    const size_t rowbase = (size_t)b * TLEN;
    const float scale = 0.08838834764831845f;   // 1/sqrt(128)

    // Stage this block's Q (64 rows x 128) into LDS via async copies.
    #pragma unroll
    for (int i = tid; i < 64 * 32; i += 128) {
        const int row = i >> 5;          // 0..63
        const int c4  = (i & 31) * 4;    // 0..124
        async_ld_b128(&Qs[row >> 4][row & 15][c4],
                      &P[(rowbase + q0 + row) * C3 + h * HS + c4]);
    }

    v8f yacc[8] = {};
    const int ktEnd = (q0 >> 4) + 4;          // causal: keys < q0+64
    const int qmax  = q0 + wave * 16 + 15;    // last q row this wave owns

    for (int kt = 0; kt < ktEnd; ++kt) {
        __syncthreads();   // prior compute's ds reads retired before overwrite
        // Stage K tile (async) and transposed V tile (16 rows x 128 each).
        #pragma unroll
        for (int i = tid; i < 16 * 32; i += 128) {
            const int row = i >> 5;
            const int c4  = (i & 31) * 4;
            const size_t r = (rowbase + kt * 16 + row) * C3 + h * HS + c4;
            async_ld_b128(&Ks[row][c4], &P[r + C_EMBD]);                  // K
            float4 v = *(const float4*)&P[r + 2 * C_EMBD];                // V
            Vt[c4 + 0][row] = v.x;
            Vt[c4 + 1][row] = v.y;
            Vt[c4 + 2][row] = v.z;
            Vt[c4 + 3][row] = v.w;
        }
        wait_async0();     // Q (first iter) + K async writes have landed
        __syncthreads();

        if (kt * 16 <= qmax) {   // wave-uniform: EXEC stays all-ones for WMMA
            // ---- S = Q * K^T : 16x16, K-dim = 128 ----
            v8f s = {};
            #pragma unroll
            for (int kk = 0; kk < HS; kk += 4) {
                v2f a = *(const v2f*)&Qs[wave][l16][kk + 2 * sub];
                v2f kb = *(const v2f*)&Ks[l16][kk + 2 * sub];   // B[d][n]=K[n][d]
                s = __builtin_amdgcn_wmma_f32_16x16x4_f32(
                    false, a, false, kb, (short)0, s, false, false);
            }
            // ---- scale, causal mask, ReLU; D-layout -> LDS row-major ----
            #pragma unroll
            for (int vr = 0; vr < 8; ++vr) {
                const int tq = q0 + wave * 16 + vr + 8 * sub;
                const int tk = kt * 16 + l16;
                float v = s[vr] * scale;
                v = (tk <= tq && v > 0.0f) ? v : 0.0f;
                Ss[wave][vr + 8 * sub][l16] = v;
            }
            // same-wave ds store->load stays in order (DScnt in-order)

            // ---- Y += S * V : 16x128, K-dim = 16 ----
            #pragma unroll
            for (int kb = 0; kb < 16; kb += 4) {
                v2f a = *(const v2f*)&Ss[wave][l16][kb + 2 * sub];
                #pragma unroll
                for (int nt = 0; nt < 8; ++nt) {
                    // B[k][n] = V[key k][col n] = Vt[n][k] -> contiguous pair
                    v2f vb = *(const v2f*)&Vt[nt * 16 + l16][kb + 2 * sub];
                    yacc[nt] = __builtin_amdgcn_wmma_f32_16x16x4_f32(
                        false, a, false, vb, (short)0, yacc[nt], false, false);
                }
            }
        }
    }

    // Store Y: y[b][t][h*128+d]
    #pragma unroll
    for (int nt = 0; nt < 8; ++nt) {
        const int d = nt * 16 + l16;
        #pragma unroll
        for (int vr = 0; vr < 8; ++vr) {
            const int t = q0 + wave * 16 + vr + 8 * sub;
            Y[(rowbase + t) * C_EMBD + h * HS + d] = yacc[nt][vr];
        }
    }
}

// ---------------------------------------------------------------------------
extern "C" void kernel_launch(void* const* d_in, const int* in_sizes, int n_in,
                              void* d_out, int out_size, void* d_ws, size_t ws_size,
                              hipStream_t stream)
{
    const float* x    = (const float*)d_in[0];   // (B,T,C)
    const float* W    = (const float*)d_in[1];   // (3C,C)
    const float* bias = (const float*)d_in[2];   // (3C)
    float* y    = (float*)d_out;                 // (B,T,C)
    float* proj = (float*)d_ws;                  // (M,3C) scratch, ~100.7 MB

    dim3 g1(MROWS / 128, C3 / 64);               // 32 x 96 blocks
    qkv_gemm_wmma<<<g1, 128, 0, stream>>>(x, W, bias, proj);

    dim3 g2(TLEN / 64, N_HEAD, BATCH);           // 32 x 16 x 2 blocks
    relu_attn_wmma<<<g2, 128, 0, stream>>>(proj, y);
}